// projector3d_48627619725591
// MI455X (gfx1250) — compile-verified
//
#include <hip/hip_runtime.h>

// Cone-beam flat-panel forward projector for MI455X (gfx1250, wave32).
// One thread = one ray (view,row,col). Gather-bound: volume (64MB) is fully
// L2-resident (192MB L2); optimize gather count + latency hiding, not FLOPs.

#define WDIM  256
#define HDIM  256
#define DDIM  256
#define NCOL  384
#define NROW  64
#define NVIEW 16
#define NSAMP 256

// RAY_LEN = sqrt(3)*256 = 443.40500673763834 ; DT = RAY_LEN/256
#define DT_F   1.7320508075688772f
// T0 = ISO - RAY_LEN/2 + 0.5*DT = 279.16352203496527
#define T0_F   279.16352203496527f

// 8-byte pair with only 4-byte alignment guarantee (x and x+1 voxels).
struct f2pair { float a, b; };

// ---------------------------------------------------------------------------
// Tiny precompute: per-view cos/sin (full ocml precision, 16 lanes total).
// Keeps the ~600-instruction Payne-Hanek range reduction OUT of the ray kernel.
// ---------------------------------------------------------------------------
__global__ void view_trig_kernel(const float* __restrict__ angles,
                                 float* __restrict__ cs)
{
    const int i = threadIdx.x;
    if (i < NVIEW) {
        const float b = angles[i];
        cs[2 * i + 0] = cosf(b);
        cs[2 * i + 1] = sinf(b);
    }
}

// ---------------------------------------------------------------------------
// Main ray kernel.
// ---------------------------------------------------------------------------
__global__ __launch_bounds__(256)
void cone_fp_kernel(const float* __restrict__ vol,
                    const float* __restrict__ cs,
                    float* __restrict__ out)
{
    const int tid = blockIdx.x * blockDim.x + threadIdx.x;
    if (tid >= NVIEW * NROW * NCOL) return;

    const int col  = tid % NCOL;
    const int rv   = tid / NCOL;
    const int row  = rv % NROW;
    const int view = rv / NROW;

    const float cb = cs[2 * view + 0];
    const float sb = cs[2 * view + 1];

    const float u = ((float)col - 191.5f) * 1.2f;   // (col - NCOL/2 + 0.5)*DET_U
    const float v = ((float)row - 31.5f) * 1.2f;    // (row - NROW/2 + 0.5)*DET_V

    // source (ISO=500) and unnormalized ray dir (det - src); |d|^2 = SDD^2+u^2+v^2
    const float sx = 500.0f * cb;
    const float sy = 500.0f * sb;
    const float dx = -1000.0f * cb - u * sb;
    const float dy = -1000.0f * sb + u * cb;
    const float dz = v;
    const float invL = 1.0f / sqrtf(fmaf(u, u, fmaf(v, v, 1.0e6f)));
    const float dirx = dx * invL, diry = dy * invL, dirz = dz * invL;

    // ray origin in voxel-index space (PIX=1, +127.5 center shift)
    const float oxv = sx + 127.5f;
    const float oyv = sy + 127.5f;
    const float ozv = 127.5f;

    // Slab-clip against trilinear support box (-1, 256)^3 in voxel coords.
    float tmin = -3.0e30f, tmax = 3.0e30f;
    {
        const float id = 1.0f / dirx;
        const float t1 = (-1.0f - oxv) * id, t2 = (256.0f - oxv) * id;
        tmin = fmaxf(tmin, fminf(t1, t2)); tmax = fminf(tmax, fmaxf(t1, t2));
    }
    {
        const float id = 1.0f / diry;
        const float t1 = (-1.0f - oyv) * id, t2 = (256.0f - oyv) * id;
        tmin = fmaxf(tmin, fminf(t1, t2)); tmax = fminf(tmax, fmaxf(t1, t2));
    }
    {
        const float id = 1.0f / dirz;   // |dirz| >= 0.6/1155 > 0, always finite
        const float t1 = (-1.0f - ozv) * id, t2 = (256.0f - ozv) * id;
        tmin = fmaxf(tmin, fminf(t1, t2)); tmax = fminf(tmax, fmaxf(t1, t2));
    }

    int kmin = 1, kmax = 0;                          // empty by default
    if (tmax > tmin) {
        kmin = (int)fmaxf(0.0f, floorf((tmin - T0_F) * (1.0f / DT_F)));
        kmax = (int)fminf((float)(NSAMP - 1), ceilf((tmax - T0_F) * (1.0f / DT_F)));
    }

    // sample position at k: p = b + k*st   (all in voxel-index coords)
    const float bx = fmaf(T0_F, dirx, oxv);
    const float by = fmaf(T0_F, diry, oyv);
    const float bz = fmaf(T0_F, dirz, ozv);
    const float stx = DT_F * dirx, sty = DT_F * diry, stz = DT_F * dirz;

    // Per-thread constant flat-index delta per sample (|st| <= 1.74 voxels):
    // used for software prefetch along the ray (L2-hit latency hiding).
    const int pfd = (int)stx + WDIM * (int)sty + (WDIM * HDIM) * (int)stz;

    float acc = 0.0f;
#pragma unroll 2
    for (int k = kmin; k <= kmax; ++k) {
        const float kf = (float)k;
        const float px = fmaf(kf, stx, bx);
        const float py = fmaf(kf, sty, by);
        const float pz = fmaf(kf, stz, bz);
        const float fx = floorf(px), fy = floorf(py), fz = floorf(pz);
        const int ix = (int)fx, iy = (int)fy, iz = (int)fz;
        const float wx = px - fx, wy = py - fy, wz = pz - fz;

        const bool interior = ((unsigned)ix < (unsigned)(WDIM - 1)) &
                              ((unsigned)iy < (unsigned)(HDIM - 1)) &
                              ((unsigned)iz < (unsigned)(DDIM - 1));
        if (interior) {
            const float* p = vol + ((iz * HDIM + iy) * WDIM + ix);
            __builtin_prefetch(p + 6 * pfd, 0, 0);   // gfx1250 global_prefetch_b8
            // 4 paired loads (x0,x0+1 contiguous) instead of 8 scalar gathers
            const f2pair a00 = *(const f2pair*)(p);
            const f2pair a01 = *(const f2pair*)(p + WDIM);
            const f2pair a10 = *(const f2pair*)(p + WDIM * HDIM);
            const f2pair a11 = *(const f2pair*)(p + WDIM * HDIM + WDIM);
            const float c00 = fmaf(wx, a00.b - a00.a, a00.a);
            const float c01 = fmaf(wx, a01.b - a01.a, a01.a);
            const float c10 = fmaf(wx, a10.b - a10.a, a10.a);
            const float c11 = fmaf(wx, a11.b - a11.a, a11.a);
            const float c0  = fmaf(wy, c01 - c00, c00);
            const float c1  = fmaf(wy, c11 - c10, c10);
            acc += fmaf(wz, c1 - c0, c0);
        } else if ((ix >= -1) & (ix <= WDIM - 1) &
                   (iy >= -1) & (iy <= HDIM - 1) &
                   (iz >= -1) & (iz <= DDIM - 1)) {
            // boundary shell: 8 guarded corners, zero outside (matches reference)
            float s = 0.0f;
#pragma unroll
            for (int dzi = 0; dzi < 2; ++dzi) {
                const int zz = iz + dzi;
                if ((unsigned)zz >= (unsigned)DDIM) continue;
                const float wzz = dzi ? wz : 1.0f - wz;
#pragma unroll
                for (int dyi = 0; dyi < 2; ++dyi) {
                    const int yy = iy + dyi;
                    if ((unsigned)yy >= (unsigned)HDIM) continue;
                    const float wyy = dyi ? wy : 1.0f - wy;
#pragma unroll
                    for (int dxi = 0; dxi < 2; ++dxi) {
                        const int xx = ix + dxi;
                        if ((unsigned)xx >= (unsigned)WDIM) continue;
                        const float wxx = dxi ? wx : 1.0f - wx;
                        s = fmaf(wzz * wyy * wxx,
                                 vol[(zz * HDIM + yy) * WDIM + xx], s);
                    }
                }
            }
            acc += s;
        }
        // fully-outside samples contribute exactly 0 (already clipped anyway)
    }

    // out layout [V,NROW,NCOL] == tid under this mapping; coalesced b32 store
    out[tid] = acc * DT_F;
}

extern "C" void kernel_launch(void* const* d_in, const int* in_sizes, int n_in,
                              void* d_out, int out_size, void* d_ws, size_t ws_size,
                              hipStream_t stream) {
    (void)in_sizes; (void)n_in; (void)out_size; (void)ws_size;
    const float* vol    = (const float*)d_in[0];  // [1,1,256,256,256] f32
    const float* angles = (const float*)d_in[1];  // [16] f32
    float* out          = (float*)d_out;          // [1,1,16,64,384] f32
    float* cs           = (float*)d_ws;           // 32 floats: {cos,sin} per view

    // 1) per-view trig (16 lanes, full precision), deterministic every call
    hipLaunchKernelGGL(view_trig_kernel, dim3(1), dim3(32), 0, stream, angles, cs);

    // 2) ray casting
    const int total = NVIEW * NROW * NCOL;        // 393216 = 1536 * 256
    dim3 block(256);                               // 8 wave32 waves per block
    dim3 grid((total + 255) / 256);
    hipLaunchKernelGGL(cone_fp_kernel, grid, block, 0, stream, vol, cs, out);
}